// GDN_32504312496804
// MI455X (gfx1250) — compile-verified
//
#include <hip/hip_runtime.h>
#include <math.h>

#define N_NODES 10000
#define N_EDGES 320000
#define D 256
#define NEG_SLOPE 0.2f

typedef float v2f __attribute__((ext_vector_type(2)));
typedef float v4f __attribute__((ext_vector_type(4)));
typedef float v8f __attribute__((ext_vector_type(8)));

// Monotonic int encoding of f32: f1 < f2  <=>  enc(f1) < enc(f2) (signed int).
__device__ __forceinline__ int enc_f32(float f) {
  int i = __float_as_int(f);
  return i >= 0 ? i : (i ^ 0x7FFFFFFF);
}
__device__ __forceinline__ float dec_f32(int i) {
  return __int_as_float(i >= 0 ? i : (i ^ 0x7FFFFFFF));
}

// ---------------------------------------------------------------------------
// Kernel 0: init workspace (agg accumulator = 0, segment max = enc(-inf),
// segment denom = 0). Required every call: harness does not re-zero ws.
// ---------------------------------------------------------------------------
__global__ void gat_init(float* __restrict__ agg, int* __restrict__ menc,
                         float* __restrict__ denom) {
  int idx = blockIdx.x * blockDim.x + threadIdx.x;
  if (idx < N_NODES * D) agg[idx] = 0.0f;
  if (idx < N_NODES) {
    menc[idx] = (int)0x807FFFFF; // enc(-inf)
    denom[idx] = 0.0f;
  }
}

// ---------------------------------------------------------------------------
// Kernel 1: dual fp32 WMMA GEMM: Ps = X*Ws^T, Pt = X*Wt^T.
// One wave computes a 16x64 strip of BOTH outputs. Software-pipelined:
// the next k-step's A + 8 B fragments are loaded into fresh registers
// before the current step's 8 WMMAs execute, so VMEM latency is hidden
// behind matrix math instead of serialized s_wait_loadcnt 0 stalls.
// 4 waves/block share mtile (A rows hit L0), nquad strides the N dim.
// ---------------------------------------------------------------------------
__global__ __launch_bounds__(128) void gat_proj(
    const float* __restrict__ X, const float* __restrict__ Ws,
    const float* __restrict__ Wt, float* __restrict__ Ps,
    float* __restrict__ Pt) {
  const int lane = threadIdx.x & 31;
  const int wave = blockIdx.x * 4 + (threadIdx.x >> 5);
  const int mtile = wave >> 2;
  const int nquad = wave & 3;
  const int mbase = mtile * 16;
  const int nbase = nquad * 64;
  const int r  = lane & 15;          // A-row / B-col within tile
  const int kk = (lane >> 4) * 2;    // k sub-offset within a 4-step

  v8f accS[4] = {};
  v8f accT[4] = {};

  const float* aptr = X  + (size_t)(mbase + r) * D + kk;
  const float* bsp  = Ws + (size_t)(nbase + r) * D + kk;
  const float* btp  = Wt + (size_t)(nbase + r) * D + kk;

  // Prologue: fragments for k-step 0.
  v2f a = *(const v2f*)(aptr);
  v2f bs[4], bt[4];
#pragma unroll
  for (int t = 0; t < 4; ++t) {
    bs[t] = *(const v2f*)(bsp + t * 16 * D);
    bt[t] = *(const v2f*)(btp + t * 16 * D);
  }

#pragma unroll 4
  for (int k0 = 0; k0 < D - 4; k0 += 4) {
    // Prefetch next k-step into fresh registers (9 loads in flight).
    v2f an = *(const v2f*)(aptr + k0 + 4);
    v2f bsn[4], btn[4];
#pragma unroll
    for (int t = 0; t < 4; ++t) {
      bsn[t] = *(const v2f*)(bsp + t * 16 * D + k0 + 4);
      btn[t] = *(const v2f*)(btp + t * 16 * D + k0 + 4);
    }
    // Compute current k-step: 8 independent WMMAs cover the load latency.
#pragma unroll
    for (int t = 0; t < 4; ++t) {
      accS[t] = __builtin_amdgcn_wmma_f32_16x16x4_f32(
          false, a, false, bs[t], (short)0, accS[t], false, false);
      accT[t] = __builtin_amdgcn_wmma_f32_16x16x4_f32(
          false, a, false, bt[t], (short)0, accT[t], false, false);
    }
    a = an;
#pragma unroll
    for (int t = 0; t < 4; ++t) { bs[t] = bsn[t]; bt[t] = btn[t]; }
  }
  // Epilogue: last k-step.
#pragma unroll
  for (int t = 0; t < 4; ++t) {
    accS[t] = __builtin_amdgcn_wmma_f32_16x16x4_f32(
        false, a, false, bs[t], (short)0, accS[t], false, false);
    accT[t] = __builtin_amdgcn_wmma_f32_16x16x4_f32(
        false, a, false, bt[t], (short)0, accT[t], false, false);
  }

  const int mrow = 8 * (lane >> 4);
#pragma unroll
  for (int t = 0; t < 4; ++t) {
#pragma unroll
    for (int j = 0; j < 8; ++j) {
      int row = mbase + mrow + j;
      int col = nbase + t * 16 + r;
      Ps[(size_t)row * D + col] = accS[t][j];
      Pt[(size_t)row * D + col] = accT[t][j];
    }
  }
}

// ---------------------------------------------------------------------------
// Kernel 2: per-edge logit e = W_attn . LeakyReLU(Ps[src] + Pt[dst])
// One wave per edge (32 lanes x 8 f32 = 256). Rows are L2-resident (10 MB).
// Segment max via int-encoded atomicMax.
// ---------------------------------------------------------------------------
__global__ __launch_bounds__(256) void gat_edge_logit(
    const float* __restrict__ Ps, const float* __restrict__ Pt,
    const int* __restrict__ src, const int* __restrict__ dst,
    const float* __restrict__ Wa, float* __restrict__ e,
    int* __restrict__ menc) {
  int edge = blockIdx.x * 8 + (threadIdx.x >> 5);
  if (edge >= N_EDGES) return;
  int lane = threadIdx.x & 31;
  int s = src[edge];
  int d = dst[edge];
  const float* ps = Ps + (size_t)s * D + lane * 8;
  const float* pt = Pt + (size_t)d * D + lane * 8;
  const float* wa = Wa + lane * 8;
  float acc = 0.0f;
#pragma unroll
  for (int j = 0; j < 8; j += 4) {
    v4f a = *(const v4f*)(ps + j);
    v4f b = *(const v4f*)(pt + j);
    v4f w = *(const v4f*)(wa + j);
#pragma unroll
    for (int c = 0; c < 4; ++c) {
      float z = a[c] + b[c];
      z = (z >= 0.0f) ? z : NEG_SLOPE * z;
      acc += z * w[c];
    }
  }
#pragma unroll
  for (int off = 16; off > 0; off >>= 1) acc += __shfl_xor(acc, off, 32);
  if (lane == 0) {
    e[edge] = acc;
    atomicMax(menc + s, enc_f32(acc));
  }
}

// ---------------------------------------------------------------------------
// Kernel 3: e_exp = exp(e - m[src]); denom[src] += e_exp (f32 atomic).
// ---------------------------------------------------------------------------
__global__ void gat_edge_exp(const float* __restrict__ e,
                             const int* __restrict__ src,
                             const int* __restrict__ menc,
                             float* __restrict__ eexp,
                             float* __restrict__ denom) {
  int i = blockIdx.x * blockDim.x + threadIdx.x;
  if (i >= N_EDGES) return;
  int s = src[i];
  float m = dec_f32(menc[s]);
  float v = expf(e[i] - m);
  eexp[i] = v;
  unsafeAtomicAdd(denom + s, v);
}

// ---------------------------------------------------------------------------
// Kernel 4: agg[dst] += feature[src] * alpha. One wave per edge; agg (10 MB)
// is L2-resident, scatter uses global_atomic_add_f32.
// ---------------------------------------------------------------------------
__global__ __launch_bounds__(256) void gat_scatter(
    const float* __restrict__ X, const int* __restrict__ src,
    const int* __restrict__ dst, const float* __restrict__ eexp,
    const float* __restrict__ denom, float* __restrict__ agg) {
  int edge = blockIdx.x * 8 + (threadIdx.x >> 5);
  if (edge >= N_EDGES) return;
  int lane = threadIdx.x & 31;
  int s = src[edge];
  int d = dst[edge];
  float alpha = eexp[edge] / denom[s];
  const float* x = X + (size_t)s * D + lane * 8;
  float* a = agg + (size_t)d * D + lane * 8;
#pragma unroll
  for (int j = 0; j < 8; j += 4) {
    v4f v = *(const v4f*)(x + j);
#pragma unroll
    for (int c = 0; c < 4; ++c) unsafeAtomicAdd(a + j + c, v[c] * alpha);
  }
}

// ---------------------------------------------------------------------------
// Kernel 5: out = agg * W_fc^T + b_fc (fp32 WMMA, software-pipelined like
// gat_proj but single weight matrix; bias added at store).
// ---------------------------------------------------------------------------
__global__ __launch_bounds__(128) void gat_out(
    const float* __restrict__ A, const float* __restrict__ W,
    const float* __restrict__ bias, float* __restrict__ out) {
  const int lane = threadIdx.x & 31;
  const int wave = blockIdx.x * 4 + (threadIdx.x >> 5);
  const int mtile = wave >> 2;
  const int nquad = wave & 3;
  const int mbase = mtile * 16;
  const int nbase = nquad * 64;
  const int r  = lane & 15;
  const int kk = (lane >> 4) * 2;

  v8f acc[4] = {};
  const float* aptr = A + (size_t)(mbase + r) * D + kk;
  const float* bp   = W + (size_t)(nbase + r) * D + kk;

  v2f a = *(const v2f*)(aptr);
  v2f b[4];
#pragma unroll
  for (int t = 0; t < 4; ++t) b[t] = *(const v2f*)(bp + t * 16 * D);

#pragma unroll 4
  for (int k0 = 0; k0 < D - 4; k0 += 4) {
    v2f an = *(const v2f*)(aptr + k0 + 4);
    v2f bn[4];
#pragma unroll
    for (int t = 0; t < 4; ++t) bn[t] = *(const v2f*)(bp + t * 16 * D + k0 + 4);
#pragma unroll
    for (int t = 0; t < 4; ++t)
      acc[t] = __builtin_amdgcn_wmma_f32_16x16x4_f32(
          false, a, false, b[t], (short)0, acc[t], false, false);
    a = an;
#pragma unroll
    for (int t = 0; t < 4; ++t) b[t] = bn[t];
  }
#pragma unroll
  for (int t = 0; t < 4; ++t)
    acc[t] = __builtin_amdgcn_wmma_f32_16x16x4_f32(
        false, a, false, b[t], (short)0, acc[t], false, false);

  const int mrow = 8 * (lane >> 4);
#pragma unroll
  for (int t = 0; t < 4; ++t) {
    float bb = bias[nbase + t * 16 + r];
#pragma unroll
    for (int j = 0; j < 8; ++j) {
      int row = mbase + mrow + j;
      int col = nbase + t * 16 + r;
      out[(size_t)row * D + col] = acc[t][j] + bb;
    }
  }
}

// ---------------------------------------------------------------------------
extern "C" void kernel_launch(void* const* d_in, const int* in_sizes, int n_in,
                              void* d_out, int out_size, void* d_ws,
                              size_t ws_size, hipStream_t stream) {
  const float* feature = (const float*)d_in[0];
  const int*   src     = (const int*)d_in[1];
  const int*   dst     = (const int*)d_in[2];
  const float* W_s     = (const float*)d_in[3];
  const float* W_t     = (const float*)d_in[4];
  const float* W_attn  = (const float*)d_in[5];
  const float* W_fc    = (const float*)d_in[6];
  const float* b_fc    = (const float*)d_in[7];
  float* out = (float*)d_out;

  // Workspace layout (bytes)
  char* ws = (char*)d_ws;
  float* Ps    = (float*)(ws);                                   // N*D f32
  float* Pt    = (float*)(ws + (size_t)N_NODES * D * 4);         // N*D f32
  float* agg   = (float*)(ws + (size_t)N_NODES * D * 8);         // N*D f32
  float* e     = (float*)(ws + (size_t)N_NODES * D * 12);        // E f32
  float* eexp  = (float*)(e + N_EDGES);                          // E f32
  int*   menc  = (int*)  (eexp + N_EDGES);                       // N i32
  float* denom = (float*)(menc + N_NODES);                       // N f32

  // 0: init accumulators
  gat_init<<<(N_NODES * D + 255) / 256, 256, 0, stream>>>(agg, menc, denom);

  // 1: projection GEMMs (625 blocks x 4 waves; waves share mtile)
  gat_proj<<<N_NODES / 16, 128, 0, stream>>>(feature, W_s, W_t, Ps, Pt);

  // 2: edge logits + segment max (8 edges per 256-thread block)
  gat_edge_logit<<<N_EDGES / 8, 256, 0, stream>>>(Ps, Pt, src, dst, W_attn, e, menc);

  // 3: exp + segment sum
  gat_edge_exp<<<(N_EDGES + 255) / 256, 256, 0, stream>>>(e, src, menc, eexp, denom);

  // 4: weighted scatter to agg
  gat_scatter<<<N_EDGES / 8, 256, 0, stream>>>(feature, src, dst, eexp, denom, agg);

  // 5: final GEMM + bias
  gat_out<<<N_NODES / 16, 128, 0, stream>>>(agg, W_fc, b_fc, out);
}